// SlidingWindowAttention_87445534146711
// MI455X (gfx1250) — compile-verified
//
#include <hip/hip_runtime.h>

// Sliding-window causal attention masking for MI455X (gfx1250).
//
// out[b,h,r,c] = scores[b,h,r,c]  if (c <= r) && (r - c <= WINDOW)
//              = -1e9             otherwise
//
// Pure streaming op (0 FLOP/byte): optimized purely for bytes moved.
//   reads : only the in-window band (<=257 of 2048 cols/row) ~ 50 MB
//   writes: full tensor (harness requires it)               ~ 403 MB
//   => ~453 MB @ 23.3 TB/s ~ 19.4 us (vs 34.6 us for naive read-all).
//
// CDNA5 paths used:
//   - constant-fill region (87.5% of output) is streamed with
//     GLOBAL_STORE_ASYNC_FROM_LDS_B128 (no VGPR payload, ASYNCcnt-tracked),
//     drained with s_wait_asynccnt 0. Falls back to NT b128 stores if the
//     builtin is unavailable.
//   - windowed data path uses non-temporal b128 loads/stores (read-once /
//     write-once streams; no reuse worth keeping in the 192 MB L2).

typedef __attribute__((ext_vector_type(4))) float float4v;
// Exact pointee type expected by the async builtin (per hipcc diagnostic):
// "__attribute__((__vector_size__(4 * sizeof(int)))) int __device__ *"
typedef int b128_t __attribute__((vector_size(16)));

#define SWA_WINDOW   256u
#define SWA_NEG_INF  (-1000000000.0f)

#if defined(__has_builtin)
#if __has_builtin(__builtin_amdgcn_global_store_async_from_lds_b128) && \
    __has_builtin(__builtin_amdgcn_s_wait_asynccnt)
#define SWA_ASYNC_FILL 1
#endif
#endif

__global__ __launch_bounds__(256) void swa_mask_kernel(
    const float* __restrict__ scores,
    const int*   __restrict__ seq_len_p,
    float*       __restrict__ out,
    unsigned int n4)                       // total number of float4 elements
{
#ifdef SWA_ASYNC_FILL
    // 16-byte constant block in LDS; broadcast-read by every lane of the
    // async fill stores (conflict-free).
    __shared__ __align__(16) float lds_neg[4];
    if (threadIdx.x < 4u) lds_neg[threadIdx.x] = SWA_NEG_INF;
    __syncthreads();   // before any early return: all threads reach this
#endif

    // Uniform across the grid: s_load + scalarized reciprocal divides.
    const unsigned int T  = (unsigned int)*seq_len_p;
    const unsigned int T4 = T >> 2;        // float4s per row (T % 4 == 0)

    const unsigned int idx4 = blockIdx.x * blockDim.x + threadIdx.x;
    if (idx4 < n4) {
        // row in [0, B*H*T); r = row within the T x T score tile.
        const unsigned int row = idx4 / T4;
        const unsigned int c0  = (idx4 - row * T4) * 4u; // first col of this float4
        const unsigned int r   = row % T;

        // Kept columns: c in [lo, hi] inclusive (causal sliding window).
        const unsigned int lo = (r >= SWA_WINDOW) ? (r - SWA_WINDOW) : 0u;
        const unsigned int hi = r;

        const float4v NEG4 = { SWA_NEG_INF, SWA_NEG_INF, SWA_NEG_INF, SWA_NEG_INF };

        if (c0 + 3u < lo || c0 > hi) {
            // Fully masked: constant fill, no global load issued.
#ifdef SWA_ASYNC_FILL
            __builtin_amdgcn_global_store_async_from_lds_b128(
                (__attribute__((address_space(1))) b128_t*)((float4v*)out + idx4),
                (__attribute__((address_space(3))) b128_t*)lds_neg,
                /*offset=*/0, /*cpol=*/0);
#else
            __builtin_nontemporal_store(NEG4, (float4v*)out + idx4);
#endif
        } else {
            const float4v* src = (const float4v*)scores + idx4;
            float4v s = __builtin_nontemporal_load(src);   // b128 NT load
            float4v v;
            if (c0 >= lo && c0 + 3u <= hi) {
                v = s;                                     // fully inside window
            } else {
                v = NEG4;                                  // straddles an edge
#pragma unroll
                for (int k = 0; k < 4; ++k) {
                    const unsigned int c = c0 + (unsigned int)k;
                    if (c >= lo && c <= hi) v[k] = s[k];
                }
            }
            __builtin_nontemporal_store(v, (float4v*)out + idx4); // b128 NT store
        }
    }

#ifdef SWA_ASYNC_FILL
    // Explicit drain of the async fill stores (endpgm's implicit S_WAIT_IDLE
    // would also cover this; keep it explicit for clarity).
    __builtin_amdgcn_s_wait_asynccnt(0);
#endif
}

extern "C" void kernel_launch(void* const* d_in, const int* in_sizes, int n_in,
                              void* d_out, int out_size, void* d_ws, size_t ws_size,
                              hipStream_t stream) {
    const float* scores  = (const float*)d_in[0];   // (B, H, T, T) fp32
    const int*   seq_len = (const int*)d_in[1];     // scalar T on device
    float*       out     = (float*)d_out;

    const unsigned int n4     = (unsigned int)(out_size / 4); // T % 4 == 0
    const unsigned int tpb    = 256;                          // 8 wave32 per block
    const unsigned int blocks = (n4 + tpb - 1) / tpb;

    hipLaunchKernelGGL(swa_mask_kernel, dim3(blocks), dim3(tpb), 0, stream,
                       scores, seq_len, out, n4);
}